// GAT_layer_59219009077249
// MI455X (gfx1250) — compile-verified
//
#include <hip/hip_runtime.h>

#define NN 8192
#define IN_F 512
#define OUT_F 64
#define ALPHA 0.2f
#define NEG_INF -1000000000000.0f

typedef __attribute__((ext_vector_type(2))) float v2f;
typedef __attribute__((ext_vector_type(8))) float v8f;

__device__ __forceinline__ v8f wmma_f32x4(v2f a, v2f b, v8f c) {
    // D = A(16x4 f32) * B(4x16 f32) + C(16x16 f32)
    return __builtin_amdgcn_wmma_f32_16x16x4_f32(
        /*neg_a=*/false, a, /*neg_b=*/false, b,
        /*c_mod=*/(short)0, c, /*reuse_a=*/false, /*reuse_b=*/false);
}

// ---------------------------------------------------------------------------
// Kernel 1: h = X @ W (8192x512 @ 512x64) via fp32 WMMA 16x16x4.
// One wave computes a 16x64 tile. Also writes hT (64x8192, col-major copy)
// so the attention kernel can load B fragments as contiguous b64 pairs.
// A-layout: lane L -> row L%16, VGPR v -> K = 2*(L/16)+v.
// B-layout: lane L -> col L%16, VGPR v -> K = 2*(L/16)+v.
// C-layout: lane L, VGPR v -> row v + 8*(L/16), col L%16.
// ---------------------------------------------------------------------------
__global__ __launch_bounds__(128) void gemm_h_kernel(
    const float* __restrict__ X, const float* __restrict__ W,
    float* __restrict__ h, float* __restrict__ hT) {
    const int lane = threadIdx.x & 31;
    const int wv   = threadIdx.x >> 5;
    const int tile = blockIdx.x * 4 + wv;
    const int i0   = tile * 16;
    const int r    = lane & 15;
    const int hf   = lane >> 4;

    v8f acc[4] = {};
    const float* xrow = X + (size_t)(i0 + r) * IN_F + 2 * hf;
    for (int k = 0; k < IN_F; k += 4) {
        v2f a = *(const v2f*)(xrow + k);
        const float* w0 = W + (size_t)(k + 2 * hf) * OUT_F + r;
#pragma unroll
        for (int t = 0; t < 4; ++t) {
            v2f b;
            b.x = w0[t * 16];
            b.y = w0[OUT_F + t * 16];
            acc[t] = wmma_f32x4(a, b, acc[t]);
        }
    }
#pragma unroll
    for (int t = 0; t < 4; ++t)
#pragma unroll
        for (int v = 0; v < 8; ++v) {
            const int row = i0 + 8 * hf + v;
            const int col = t * 16 + r;
            h[(size_t)row * OUT_F + col] = acc[t][v];
            hT[(size_t)col * NN + row]  = acc[t][v];
        }
}

// ---------------------------------------------------------------------------
// Kernel 2: f1[i] = h[i,:].a[:64], f2[i] = h[i,:].a[64:128]
// ---------------------------------------------------------------------------
__global__ __launch_bounds__(128) void proj_f_kernel(
    const float* __restrict__ h, const float* __restrict__ a,
    float* __restrict__ f1, float* __restrict__ f2) {
    const int row = blockIdx.x * blockDim.x + threadIdx.x;
    if (row >= NN) return;
    const float* hr = h + (size_t)row * OUT_F;
    float s1 = 0.f, s2 = 0.f;
#pragma unroll 8
    for (int c = 0; c < OUT_F; ++c) {
        float v = hr[c];
        s1 += v * a[c];
        s2 += v * a[OUT_F + c];
    }
    f1[row] = s1;
    f2[row] = s2;
}

__device__ __forceinline__ float lrelu(float x) {
    return x > 0.f ? x : ALPHA * x;
}

// ---------------------------------------------------------------------------
// Kernel 3: fused masked softmax + aggregation (flash-style, 2 passes).
// One wave owns a 16-row tile. Pass 1: masked row max (16x8192 adj strip
// from DRAM). Pass 2: p = exp(s-m) built directly in WMMA A-layout, row-sum
// kept in registers, out_tile(16x64) accumulated via 16 f32 WMMAs per
// 16-col chunk; adj strip re-read hits L2; B fragments are contiguous b64
// loads from hT. Final: divide by row-sum, store.
// ---------------------------------------------------------------------------
__global__ __launch_bounds__(128) void gat_attn_kernel(
    const float* __restrict__ hT, const int* __restrict__ adj,
    const float* __restrict__ f1, const float* __restrict__ f2,
    float* __restrict__ out) {
    const int lane  = threadIdx.x & 31;
    const int wv    = threadIdx.x >> 5;
    const int tile  = blockIdx.x * 4 + wv;
    const int row0  = tile * 16;
    const int r     = lane & 15;
    const int hf    = lane >> 4;
    const int myrow = row0 + r;

    const float f1i = f1[myrow];
    const int* arow = adj + (size_t)myrow * NN;

    // ---- Pass 1: masked row max ----
    float m = NEG_INF;
    for (int j0 = 0; j0 < NN; j0 += 16) {
        const int cb = j0 + 2 * hf;
#pragma unroll
        for (int c = 0; c < 4; ++c) {
            const int col = cb + 4 * c;
            int2 av  = *(const int2*)(arow + col);
            v2f  f2p = *(const v2f*)(f2 + col);
            float s0 = (av.x > 0) ? lrelu(f1i + f2p.x) : NEG_INF;
            float s1 = (av.y > 0) ? lrelu(f1i + f2p.y) : NEG_INF;
            m = fmaxf(m, fmaxf(s0, s1));
        }
    }
    m = fmaxf(m, __shfl_xor(m, 16, 32));

    // ---- Pass 2: p = exp(s - m); accumulate l and acc += P @ h_chunk ----
    float l = 0.f;
    v8f acc[4] = {};
    for (int j0 = 0; j0 < NN; j0 += 16) {
        const int cb = j0 + 2 * hf;
        float p[8];
#pragma unroll
        for (int c = 0; c < 4; ++c) {
            const int col = cb + 4 * c;
            int2 av  = *(const int2*)(arow + col);
            v2f  f2p = *(const v2f*)(f2 + col);
            float s0 = (av.x > 0) ? lrelu(f1i + f2p.x) : NEG_INF;
            float s1 = (av.y > 0) ? lrelu(f1i + f2p.y) : NEG_INF;
            float p0 = __expf(s0 - m);
            float p1 = __expf(s1 - m);
            p[2 * c]     = p0;
            p[2 * c + 1] = p1;
            l += p0 + p1;
        }
#pragma unroll
        for (int t = 0; t < 4; ++t) {
            const float* hTcol = hT + (size_t)(t * 16 + r) * NN + cb;
#pragma unroll
            for (int c = 0; c < 4; ++c) {
                v2f A;
                A.x = p[2 * c];
                A.y = p[2 * c + 1];
                v2f B = *(const v2f*)(hTcol + 4 * c);  // contiguous K pair
                acc[t] = wmma_f32x4(A, B, acc[t]);
            }
        }
    }
    l += __shfl_xor(l, 16, 32);

    // Redistribute per-row l from A-layout (row = lane&15) to C-layout
    // (row = v + 8*(lane/16)) through LDS.
    __shared__ float lsh[4][16];
    if (hf == 0) lsh[wv][r] = l;
    __syncthreads();
    float linv[8];
#pragma unroll
    for (int v = 0; v < 8; ++v) linv[v] = 1.f / lsh[wv][8 * hf + v];
#pragma unroll
    for (int t = 0; t < 4; ++t)
#pragma unroll
        for (int v = 0; v < 8; ++v)
            out[(size_t)(row0 + 8 * hf + v) * OUT_F + t * 16 + r] =
                acc[t][v] * linv[v];
}

extern "C" void kernel_launch(void* const* d_in, const int* in_sizes, int n_in,
                              void* d_out, int out_size, void* d_ws, size_t ws_size,
                              hipStream_t stream) {
    const float* X   = (const float*)d_in[0];  // 8192x512
    const int*   adj = (const int*)d_in[1];    // 8192x8192
    const float* W   = (const float*)d_in[2];  // 512x64
    const float* a   = (const float*)d_in[3];  // 128x1
    float* out = (float*)d_out;                // 8192x64

    char*  ws = (char*)d_ws;
    float* h  = (float*)ws;                    // N*OUT_F floats (2 MB)
    float* hT = h + (size_t)NN * OUT_F;        // OUT_F*N floats (2 MB)
    float* f1 = hT + (size_t)NN * OUT_F;       // N floats
    float* f2 = f1 + NN;                       // N floats

    gemm_h_kernel<<<NN / 16 / 4, 128, 0, stream>>>(X, W, h, hT);
    proj_f_kernel<<<NN / 128, 128, 0, stream>>>(h, a, f1, f2);
    gat_attn_kernel<<<NN / 16 / 4, 128, 0, stream>>>(hT, adj, f1, f2, out);
}